// SimpleMACELayer_33509334843737
// MI455X (gfx1250) — compile-verified
//
#include <hip/hip_runtime.h>

// ---------------------------------------------------------------------------
// SimpleMACELayer on gfx1250 (MI455X), f32 WMMA 16x16x4 formulation.
//
// PATHS (reference order p=0..22), grouped by l3:
//  l3=0: p0(0,0) p5(1,1) p13(2,2) p21(3,3)
//  l3=1: p1(0,1) p4(1,0) p7(1,2) p11(2,1) p15(2,3) p19(3,2)
//  l3=2: p2(0,2) p6(1,1) p9(1,3) p10(2,0) p14(2,2) p18(3,1) p22(3,3)
//  l3=3: p3(0,3) p8(1,2) p12(2,1) p16(2,3) p17(3,0) p20(3,2)
// ---------------------------------------------------------------------------

typedef __attribute__((ext_vector_type(2))) float v2f;
typedef __attribute__((ext_vector_type(4))) float v4f;
typedef __attribute__((ext_vector_type(8))) float v8f;

#define HID 64
#define NIRR 16
#define TILE_E 16

__constant__ int GP_P[23]  = {0,5,13,21,  1,4,7,11,15,19,  2,6,9,10,14,18,22,  3,8,12,16,17,20};
__constant__ int GP_L1[23] = {0,1,2,3,    0,1,1,2,2,3,     0,1,1,2,2,3,3,      0,1,2,2,3,3};
__constant__ int GP_L2[23] = {0,1,2,3,    1,0,2,1,3,2,     2,1,3,0,2,1,3,      3,2,1,3,0,2};

static __device__ __forceinline__ v8f wmma_f32_16x16x4(v2f a, v2f b, v8f c) {
    return __builtin_amdgcn_wmma_f32_16x16x4_f32(false, a, false, b, (short)0, c, false, false);
}

// ---------------------------------------------------------------------------
// Process one l3 group: accumulate msgs[e, o, l3^2+kl] for 16 edges, then
// atomically scatter into agg[dst][o][k].
// All NK k-components of one path are built into LDS up front so the 16-step
// WMMA K-loop can interleave NK independent accumulators (breaks the RAW
// chain on the XDL pipe). NK<4 groups get a second accumulator bank
// alternating on K-step parity.
// ---------------------------------------------------------------------------
template <int L3, int G0, int NP>
static __device__ __forceinline__ void process_group(
    float (&sSrc)[TILE_E][HID][17], float (&sY)[TILE_E][NIRR],
    float (&sCg)[16][16][16], float (&sA)[TILE_E][60], float (&sU)[7][TILE_E][68],
    const int* sDst, const float* __restrict__ tpw, float* __restrict__ agg,
    int tid, int e0, int n_edges)
{
    constexpr int NK   = 2 * L3 + 1;
    constexpr int K0   = L3 * L3;
    constexpr bool DUAL = (NK < 4);
    const int lane   = tid & 31;
    const int wave   = tid >> 5;
    const int laneM  = lane & 15;   // N column (B/C) or M row (A)
    const int laneHi = lane >> 4;   // which half of the wave
    const int obase  = wave * 16;   // this wave's 16 output channels

    v8f acc[NK];
    v8f accB[DUAL ? NK : 1];
#pragma unroll
    for (int kl = 0; kl < NK; ++kl) acc[kl] = (v8f){0.f,0.f,0.f,0.f,0.f,0.f,0.f,0.f};
    if constexpr (DUAL) {
#pragma unroll
        for (int kl = 0; kl < NK; ++kl) accB[kl] = (v8f){0.f,0.f,0.f,0.f,0.f,0.f,0.f,0.f};
    }

    for (int gp = G0; gp < G0 + NP; ++gp) {
        const int p  = GP_P[gp];
        const int l1 = GP_L1[gp], l2 = GP_L2[gp];
        const int i0 = l1 * l1, ni = 2 * l1 + 1;
        const int j0 = l2 * l2, nj = 2 * l2 + 1;

        // Preload this path's B tile (W_p[h, obase..obase+15]) into registers.
        // B layout (4x16 f32): vgpr0 = rows {4s, 4s+1}, vgpr1 = rows {4s+2, 4s+3}.
        float b0[16], b1[16];
        const float* Wp = tpw + ((size_t)p * HID) * HID + obase + laneM;
#pragma unroll
        for (int s = 0; s < 16; ++s) {
            b0[s] = Wp[(4 * s + laneHi) * HID];
            b1[s] = Wp[(4 * s + 2 + laneHi) * HID];
        }

        // A_p[e, i, kl] = sum_j Y[e, j0+j] * cg[i0+i, j0+j, K0+kl], all kl at once.
        {
            const int eA = tid >> 3, il = tid & 7;
            if (il < ni) {
#pragma unroll
                for (int kl = 0; kl < NK; ++kl) {
                    float a = 0.f;
                    for (int j = 0; j < nj; ++j)
                        a += sY[eA][j0 + j] * sCg[i0 + il][j0 + j][K0 + kl];
                    sA[eA][kl * 8 + il] = a;
                }
            }
        }
        __syncthreads();

        // u[kl][e, h] = sum_i sSrc[e, h, i0+i] * A[e, i, kl]  (all NK planes)
        {
            const int h = tid & 63;
            for (int e = (tid >> 6); e < TILE_E; e += 2) {
#pragma unroll
                for (int kl = 0; kl < NK; ++kl) {
                    float u = 0.f;
                    for (int i = 0; i < ni; ++i)
                        u += sSrc[e][h][i0 + i] * sA[e][kl * 8 + i];
                    sU[kl][e][h] = u;
                }
            }
        }
        __syncthreads();

        // GEMM: acc[kl](16e x 16o) += u[kl](16e x 64h) * W_p(64h x 16o)
        // A layout (16x4 f32): lanes 0-15 hold K={4s,4s+1}, 16-31 hold {4s+2,4s+3}.
        // Inner loop over kl -> NK independent WMMA accumulator chains.
#pragma unroll
        for (int s = 0; s < 16; ++s) {
            v2f b2;
            b2.x = b0[s];
            b2.y = b1[s];
#pragma unroll
            for (int kl = 0; kl < NK; ++kl) {
                v2f a2;
                a2.x = sU[kl][laneM][4 * s + 2 * laneHi];
                a2.y = sU[kl][laneM][4 * s + 2 * laneHi + 1];
                if (DUAL && (s & 1))
                    accB[kl] = wmma_f32_16x16x4(a2, b2, accB[kl]);
                else
                    acc[kl] = wmma_f32_16x16x4(a2, b2, acc[kl]);
            }
        }
        __syncthreads();   // protect sA/sU before next path
    }

    if constexpr (DUAL) {
#pragma unroll
        for (int kl = 0; kl < NK; ++kl) acc[kl] += accB[kl];
    }

    // Scatter: C layout: vgpr r -> edge (r + 8*laneHi), col = obase+laneM.
#pragma unroll
    for (int kl = 0; kl < NK; ++kl) {
        const int k = K0 + kl;
#pragma unroll
        for (int r = 0; r < 8; ++r) {
            const int e = r + laneHi * 8;
            if (e0 + e < n_edges) {
                const int dst = sDst[e];
                atomicAdd(agg + ((size_t)dst * HID + obase + laneM) * NIRR + k, acc[kl][r]);
            }
        }
    }
}

// ---------------------------------------------------------------------------
// Kernel 1: per-edge tensor product messages + atomic scatter into agg.
// One WG (128 threads, 4 waves) per 16 edges.
// ---------------------------------------------------------------------------
__global__ __launch_bounds__(128) void edge_kernel(
    const float* __restrict__ nf,   // [A][64][16]
    const float* __restrict__ ev,   // [E][3] (unit vectors)
    const int*   __restrict__ eidx, // [2][E]
    const float* __restrict__ cg,   // [16][16][16]
    const float* __restrict__ tpw,  // [23][64][64]
    float* __restrict__ agg,        // [A][64][16]
    int n_edges)
{
    __shared__ float sSrc[TILE_E][HID][17];
    __shared__ float sY[TILE_E][NIRR];
    __shared__ float sCg[16][16][16];
    __shared__ float sA[TILE_E][60];
    __shared__ float sU[7][TILE_E][68];
    __shared__ int   sDst[TILE_E];
    __shared__ int   sSid[TILE_E];

    const int tid = threadIdx.x;
    const int e0  = blockIdx.x * TILE_E;

    // cg -> LDS (one-time, 16 KB)
    for (int w = tid; w < 4096; w += 128) ((float*)sCg)[w] = cg[w];

    if (tid < TILE_E) {
        const int e = min(e0 + tid, n_edges - 1);
        sSid[tid] = eidx[e];
        sDst[tid] = eidx[n_edges + e];
        const float x = ev[3 * e + 0], y = ev[3 * e + 1], z = ev[3 * e + 2];
        const float x2 = x * x, y2 = y * y, z2 = z * z;
        sY[tid][0]  = 0.28209479177387814f;
        sY[tid][1]  = 0.4886025119029199f * y;
        sY[tid][2]  = 0.4886025119029199f * z;
        sY[tid][3]  = 0.4886025119029199f * x;
        sY[tid][4]  = 1.0925484305920792f * x * y;
        sY[tid][5]  = 1.0925484305920792f * y * z;
        sY[tid][6]  = 0.31539156525252005f * (3.0f * z2 - 1.0f);
        sY[tid][7]  = 1.0925484305920792f * x * z;
        sY[tid][8]  = 0.5462742152960396f * (x2 - y2);
        sY[tid][9]  = 0.5900435899266435f * y * (3.0f * x2 - y2);
        sY[tid][10] = 2.890611442640554f * x * y * z;
        sY[tid][11] = 0.4570457994644658f * y * (5.0f * z2 - 1.0f);
        sY[tid][12] = 0.3731763325901154f * z * (5.0f * z2 - 3.0f);
        sY[tid][13] = 0.4570457994644658f * x * (5.0f * z2 - 1.0f);
        sY[tid][14] = 1.445305721320277f * z * (x2 - y2);
        sY[tid][15] = 0.5900435899266435f * x * (x2 - 3.0f * y2);
    }
    __syncthreads();

    // Gather src features: 16 edges x 1024 floats, b128 loads from L2.
    for (int w4 = tid; w4 < 4096; w4 += 128) {
        const int e  = w4 >> 8;
        const int r  = (w4 & 255) * 4;   // 0..1020, 16B aligned
        const v4f v  = *(const v4f*)(nf + (size_t)sSid[e] * (HID * NIRR) + r);
        const int h  = r >> 4, i = r & 15;
        sSrc[e][h][i + 0] = v.x;
        sSrc[e][h][i + 1] = v.y;
        sSrc[e][h][i + 2] = v.z;
        sSrc[e][h][i + 3] = v.w;
    }
    __syncthreads();

    process_group<0, 0, 4>(sSrc, sY, sCg, sA, sU, sDst, tpw, agg, tid, e0, n_edges);
    process_group<1, 4, 6>(sSrc, sY, sCg, sA, sU, sDst, tpw, agg, tid, e0, n_edges);
    process_group<2, 10, 7>(sSrc, sY, sCg, sA, sU, sDst, tpw, agg, tid, e0, n_edges);
    process_group<3, 17, 6>(sSrc, sY, sCg, sA, sU, sDst, tpw, agg, tid, e0, n_edges);
}

// ---------------------------------------------------------------------------
// Kernel 2: out[a, o2] = sum_c agg_flat[a, c] * W_lin[o2, c] + b_lin[o2]
// WMMA GEMM, M=16 atoms per WG, K=1024, N=64 (4 waves x 16).
// 4 interleaved accumulators to break the WMMA RAW chain.
// ---------------------------------------------------------------------------
__global__ __launch_bounds__(128) void linear_kernel(
    const float* __restrict__ agg, const float* __restrict__ Wl,
    const float* __restrict__ bl, float* __restrict__ out, int n_atoms)
{
    __shared__ float sAgg[16][1028];
    const int tid = threadIdx.x;
    const int a0  = blockIdx.x * 16;

    for (int w4 = tid; w4 < 4096; w4 += 128) {
        const int m = w4 >> 8;
        const int c = (w4 & 255) * 4;
        const int a = a0 + m;
        v4f v = (v4f){0.f, 0.f, 0.f, 0.f};
        if (a < n_atoms) v = *(const v4f*)(agg + (size_t)a * 1024 + c);
        sAgg[m][c + 0] = v.x;
        sAgg[m][c + 1] = v.y;
        sAgg[m][c + 2] = v.z;
        sAgg[m][c + 3] = v.w;
    }
    __syncthreads();

    const int lane = tid & 31, wave = tid >> 5;
    const int laneM = lane & 15, laneHi = lane >> 4;
    const int obase = wave * 16;

    v8f acc0 = (v8f){0.f,0.f,0.f,0.f,0.f,0.f,0.f,0.f};
    v8f acc1 = acc0, acc2 = acc0, acc3 = acc0;
    const float* wrow = Wl + (size_t)(obase + laneM) * 1024;  // B[c][n] = W_lin[n][c]

#pragma unroll 2
    for (int s4 = 0; s4 < 256; s4 += 4) {
#pragma unroll
        for (int q = 0; q < 4; ++q) {
            const int s = s4 + q;
            v2f a2;
            a2.x = sAgg[laneM][4 * s + 2 * laneHi];
            a2.y = sAgg[laneM][4 * s + 2 * laneHi + 1];
            v2f b2;
            b2.x = wrow[4 * s + laneHi];
            b2.y = wrow[4 * s + 2 + laneHi];
            if (q == 0)      acc0 = wmma_f32_16x16x4(a2, b2, acc0);
            else if (q == 1) acc1 = wmma_f32_16x16x4(a2, b2, acc1);
            else if (q == 2) acc2 = wmma_f32_16x16x4(a2, b2, acc2);
            else             acc3 = wmma_f32_16x16x4(a2, b2, acc3);
        }
    }
    v8f acc = (acc0 + acc1) + (acc2 + acc3);

    const float bias = bl[obase + laneM];
#pragma unroll
    for (int r = 0; r < 8; ++r) {
        const int a = a0 + r + laneHi * 8;
        if (a < n_atoms)
            out[(size_t)a * 64 + obase + laneM] = acc[r] + bias;
    }
}

__global__ void zero_kernel(float* __restrict__ p, int n4) {
    const int i = blockIdx.x * 256 + threadIdx.x;
    if (i < n4) ((v4f*)p)[i] = (v4f){0.f, 0.f, 0.f, 0.f};
}

// ---------------------------------------------------------------------------
extern "C" void kernel_launch(void* const* d_in, const int* in_sizes, int n_in,
                              void* d_out, int out_size, void* d_ws, size_t ws_size,
                              hipStream_t stream) {
    const float* nf  = (const float*)d_in[0];   // node_features  [A,64,16]
    const float* ev  = (const float*)d_in[1];   // edge_vectors   [E,3]
    const int*   ei  = (const int*)d_in[2];     // edge_index     [2,E]
    const float* cg  = (const float*)d_in[3];   // gaunt          [16,16,16]
    const float* tpw = (const float*)d_in[4];   // tp_weights     [23,64,64]
    const float* Wl  = (const float*)d_in[5];   // W_lin          [64,1024]
    const float* bl  = (const float*)d_in[6];   // b_lin          [64]

    const int n_atoms = in_sizes[0] / (HID * NIRR);
    const int n_edges = in_sizes[1] / 3;

    float* agg = (float*)d_ws;                  // [A,64,16] accumulator
    const int agg4 = n_atoms * HID * NIRR / 4;

    zero_kernel<<<(agg4 + 255) / 256, 256, 0, stream>>>(agg, agg4);
    edge_kernel<<<(n_edges + TILE_E - 1) / TILE_E, 128, 0, stream>>>(
        nf, ev, ei, cg, tpw, agg, n_edges);
    linear_kernel<<<(n_atoms + 15) / 16, 128, 0, stream>>>(
        agg, Wl, bl, (float*)d_out, n_atoms);
}